// MoEGate_6150393168540
// MI455X (gfx1250) — compile-verified
//
#include <hip/hip_runtime.h>
#include <math.h>

typedef __attribute__((ext_vector_type(2))) float v2f;
typedef __attribute__((ext_vector_type(8))) float v8f;

constexpr int kD = 1024;          // feature dim
constexpr int kE = 8;             // experts
constexpr int kLWS = kD + 4;      // LDS weight row stride (floats), pad 4 -> conflict-free b64 reads
constexpr int kSCS = 20;          // LDS scratch row stride (floats), 80B (16B aligned), low conflict

__global__ __launch_bounds__(256) void moe_gate_wmma_kernel(
    const float* __restrict__ x,
    const float* __restrict__ gate_w,
    const float* __restrict__ gate_b,
    const float* __restrict__ expert_w,
    const float* __restrict__ expert_b,
    float* __restrict__ out)
{
    // 16 weight rows (8 gate + 8 expert), padded + per-wave 32x16 C scratch
    __shared__ float smem[16 * kLWS + 8 * 32 * kSCS];
    float* lw = smem;                      // combined weights [16][kLWS]
    float* sc = smem + 16 * kLWS;          // scratch [8 waves][32 rows][kSCS]

    const int tid  = threadIdx.x;          // 0..255
    const int wave = tid >> 5;             // 0..7
    const int lane = tid & 31;             // 0..31
    const int lo   = lane & 15;            // 0..15
    const int hi   = lane >> 4;            // 0 or 1

    // ---- stage combined weight matrix into LDS (coalesced b128 per thread per row)
    {
        const float4* gw4 = (const float4*)gate_w;
        const float4* ew4 = (const float4*)expert_w;
        #pragma unroll
        for (int e = 0; e < kE; ++e) {
            float4 g = gw4[e * (kD / 4) + tid];
            *(float4*)&lw[e * kLWS + tid * 4] = g;
            float4 w = ew4[e * (kD / 4) + tid];
            *(float4*)&lw[(kE + e) * kLWS + tid * 4] = w;
        }
    }
    __syncthreads();

    // ---- this wave's 32 rows: two 16-row WMMA tiles
    const long long blk_row = (long long)blockIdx.x * 256 + wave * 32;
    const float* xr0 = x + (blk_row + lo) * kD;         // tile 0: rows +0..15
    const float* xr1 = x + (blk_row + 16 + lo) * kD;    // tile 1: rows +16..31
    const float* lwp = &lw[lo * kLWS];                  // B fragment source row

    v8f c0 = {};
    v8f c1 = {};
    const int koff = 2 * hi;   // A/B VGPR layout: lanes 0-15 hold K0,K1; lanes 16-31 hold K2,K3

    #pragma unroll 4
    for (int k = 0; k < kD; k += 4) {
        const int kk = k + koff;
        v2f a0 = *(const v2f*)(xr0 + kk);   // A tile 0 fragment (16x4 f32)
        v2f a1 = *(const v2f*)(xr1 + kk);   // A tile 1 fragment
        v2f b  = *(const v2f*)(lwp + kk);   // B fragment (4x16 f32) from LDS
        c0 = __builtin_amdgcn_wmma_f32_16x16x4_f32(false, a0, false, b, (short)0, c0, false, false);
        c1 = __builtin_amdgcn_wmma_f32_16x16x4_f32(false, a1, false, b, (short)0, c1, false, false);
    }

    // ---- spill C tiles to LDS scratch.  C layout: VGPR r, lanes 0-15 -> (M=r, N=lane),
    //      lanes 16-31 -> (M=r+8, N=lane-16)
    float* scw = &sc[wave * 32 * kSCS];
    #pragma unroll
    for (int r = 0; r < 8; ++r) {
        scw[(r + 8 * hi) * kSCS + lo]        = c0[r];
        scw[(16 + r + 8 * hi) * kSCS + lo]   = c1[r];
    }
    __syncthreads();

    // ---- per-row epilogue: lane L handles local row L (32 rows per wave)
    const float* srow = &scw[lane * kSCS];

    float g[kE], p[kE], eo[kE];
    float m = -INFINITY;
    #pragma unroll
    for (int j = 0; j < kE; ++j) {
        g[j] = srow[j] + gate_b[j];
        m = fmaxf(m, g[j]);
    }
    float s = 0.0f;
    #pragma unroll
    for (int j = 0; j < kE; ++j) {
        p[j] = __expf(g[j] - m);
        s += p[j];
    }
    const float inv = 1.0f / s;
    #pragma unroll
    for (int j = 0; j < kE; ++j) {
        p[j] *= inv;                               // softmax prob (no renorm after masking)
        eo[j] = srow[kE + j] + expert_b[j];        // expert scalar outputs
    }

    // top-2 with first-index tie-breaking (matches lax.top_k)
    int i1 = 0; float v1 = p[0];
    #pragma unroll
    for (int j = 1; j < kE; ++j) { if (p[j] > v1) { v1 = p[j]; i1 = j; } }
    int i2 = -1; float v2 = -INFINITY;
    #pragma unroll
    for (int j = 0; j < kE; ++j) { if (j != i1 && p[j] > v2) { v2 = p[j]; i2 = j; } }

    float acc = 0.0f;
    #pragma unroll
    for (int j = 0; j < kE; ++j) {
        acc += (j == i1 || j == i2) ? p[j] * eo[j] : 0.0f;
    }

    out[blk_row + lane] = acc;
}

extern "C" void kernel_launch(void* const* d_in, const int* in_sizes, int n_in,
                              void* d_out, int out_size, void* d_ws, size_t ws_size,
                              hipStream_t stream) {
    const float* x        = (const float*)d_in[0];
    const float* gate_w   = (const float*)d_in[1];
    const float* gate_b   = (const float*)d_in[2];
    const float* expert_w = (const float*)d_in[3];
    const float* expert_b = (const float*)d_in[4];
    float* out = (float*)d_out;

    const int B = in_sizes[0] / kD;        // 65536
    const int blocks = B / 256;            // 256 rows per block (8 waves x 32 rows)

    moe_gate_wmma_kernel<<<blocks, 256, 0, stream>>>(x, gate_w, gate_b,
                                                     expert_w, expert_b, out);
}